// ChamferLossKL2_14757507629943
// MI455X (gfx1250) — compile-verified
//
#include <hip/hip_runtime.h>
#include <math.h>

// ---------------------------------------------------------------------------
// Chamfer + KL loss for MI455X (gfx1250, wave32, WMMA).
//
// dist[b,x,y] = |mu_p[b,x]|^2 + |mu_g[b,y]|^2 - 2 * <mu_p[b,x], mu_g[b,y]>
// The dot-product term is an [nx x ny] fp32 GEMM with k=32 per batch, done
// with V_WMMA_F32_16X16X4_F32. Argmins over both axes are reduced per-tile
// (never materializing the 536MB distance matrix) and merged globally via
// u64 atomicMin on (sortable_float32 << 32 | index) keys.
// ---------------------------------------------------------------------------

typedef __attribute__((ext_vector_type(2))) float v2f;
typedef __attribute__((ext_vector_type(8))) float v8f;

#define DF     32      // feature dim (fixed by reference)
#define TILE_M 128     // block tile (pred rows)
#define TILE_N 128     // block tile (gt cols)
#define WAVE_M 32      // per-wave strip rows (2 x 16)
#define WAVE_N 64      // per-wave strip cols (4 x 16)

// Monotonic float->uint mapping: u64 min == (float min, then lowest index).
__device__ __forceinline__ unsigned long long packKey(float v, int idx) {
  unsigned u = __float_as_uint(v);
  u = (u & 0x80000000u) ? ~u : (u | 0x80000000u);
  return ((unsigned long long)u << 32) | (unsigned)idx;
}

__device__ __forceinline__ unsigned long long shflXorU64(unsigned long long v, int m) {
  unsigned lo = (unsigned)v, hi = (unsigned)(v >> 32);
  lo = (unsigned)__shfl_xor((int)lo, m, 32);
  hi = (unsigned)__shfl_xor((int)hi, m, 32);
  return ((unsigned long long)hi << 32) | lo;
}

// ---------------------------------------------------------------------------
// Kernel 0: initialize packed argmin slots to +inf key, zero outputs.
// ---------------------------------------------------------------------------
__global__ void init_kernel(unsigned long long* __restrict__ argx,
                            unsigned long long* __restrict__ argy,
                            float* __restrict__ out,
                            int nArgx, int nArgy, int nOut) {
  int i = blockIdx.x * blockDim.x + threadIdx.x;
  if (i < nArgx) argx[i] = ~0ull;
  if (i < nArgy) argy[i] = ~0ull;
  if (i < nOut)  out[i]  = 0.0f;
}

// ---------------------------------------------------------------------------
// Kernel 1: squared norms, one wave per point, lane == feature (DF == 32).
// ---------------------------------------------------------------------------
__global__ void norm_kernel(const float* __restrict__ mu_p,
                            const float* __restrict__ mu_g,
                            float* __restrict__ rx, float* __restrict__ ry,
                            int bs, int nx, int ny) {
  int wid  = (blockIdx.x * blockDim.x + threadIdx.x) >> 5;
  int lane = threadIdx.x & 31;
  int t1 = bs * nx;
  int tot = t1 + bs * ny;
  if (wid >= tot) return;
  float v;
  if (wid < t1) v = mu_p[(size_t)wid * DF + lane];
  else          v = mu_g[(size_t)(wid - t1) * DF + lane];
  float s = v * v;
#pragma unroll
  for (int off = 16; off > 0; off >>= 1) s += __shfl_xor(s, off, 32);
  if (lane == 0) {
    if (wid < t1) rx[wid] = s;
    else          ry[wid - t1] = s;
  }
}

// ---------------------------------------------------------------------------
// Kernel 2: WMMA distance tiles + dual argmin reduction.
//
// Fragment layout assumption for V_WMMA_F32_16X16X4_F32 (ISA 7.12.2):
//   A (16x4, MxK): lane l holds row M=l&15, k = (l>>4)*2 + {0,1} in 2 VGPRs.
//   B (4x16, KxN): symmetric for A*B^T: lane l holds col N=l&15, same k pair.
//   C/D (16x16 f32): lane l holds col N=l&15, rows M = vg + (l>>4)*8, vg=0..7.
// ---------------------------------------------------------------------------
__global__ __launch_bounds__(256)
void chamfer_argmin_wmma(const float* __restrict__ mu_p,
                         const float* __restrict__ mu_g,
                         const float* __restrict__ rx,
                         const float* __restrict__ ry,
                         unsigned long long* __restrict__ argx,   // [bs*ny]: best pred per gt
                         unsigned long long* __restrict__ argy,   // [bs*nx]: best gt per pred
                         int nx, int ny) {
  const int b    = blockIdx.z;
  const int wave = threadIdx.x >> 5;
  const int lane = threadIdx.x & 31;
  const int lr   = lane & 15;
  const int lh   = lane >> 4;
  const int wm   = wave >> 1;            // 0..3  -> 32-row strip
  const int wn   = wave & 1;             // 0..1  -> 64-col strip
  const int m0   = blockIdx.x * TILE_M + wm * WAVE_M;
  const int n0   = blockIdx.y * TILE_N + wn * WAVE_N;

  const float* A = mu_p + (size_t)b * nx * DF;
  const float* B = mu_g + (size_t)b * ny * DF;

  v8f acc[2][4] = {};

  // K = 32 swept as 8 steps of k=4.  Loads clamped (no divergence -> EXEC
  // stays all-1s, required by WMMA); OOB rows/cols neutralized via +inf norms.
#pragma unroll
  for (int kk = 0; kk < 8; ++kk) {
    const int k = kk * 4 + lh * 2;
    v2f af[2], bf[4];
#pragma unroll
    for (int ma = 0; ma < 2; ++ma) {
      int m = m0 + ma * 16 + lr; if (m >= nx) m = nx - 1;
      af[ma] = *(const v2f*)(A + (size_t)m * DF + k);
    }
#pragma unroll
    for (int na = 0; na < 4; ++na) {
      int n = n0 + na * 16 + lr; if (n >= ny) n = ny - 1;
      bf[na] = *(const v2f*)(B + (size_t)n * DF + k);
    }
#pragma unroll
    for (int ma = 0; ma < 2; ++ma)
#pragma unroll
      for (int na = 0; na < 4; ++na)
        acc[ma][na] = __builtin_amdgcn_wmma_f32_16x16x4_f32(
            false, af[ma], false, bf[na], (short)0, acc[ma][na], false, false);
  }

  // Norms for this lane's C-fragment rows/cols (+inf for OOB => never wins).
  float rxv[2][8], ryv[4];
#pragma unroll
  for (int ma = 0; ma < 2; ++ma)
#pragma unroll
    for (int vg = 0; vg < 8; ++vg) {
      int m = m0 + ma * 16 + lh * 8 + vg;
      rxv[ma][vg] = (m < nx) ? rx[(size_t)b * nx + m] : INFINITY;
    }
#pragma unroll
  for (int na = 0; na < 4; ++na) {
    int n = n0 + na * 16 + lr;
    ryv[na] = (n < ny) ? ry[(size_t)b * ny + n] : INFINITY;
  }

  // dist = rx + ry - 2*zz (overwrite accumulators in place).
#pragma unroll
  for (int ma = 0; ma < 2; ++ma)
#pragma unroll
    for (int na = 0; na < 4; ++na)
#pragma unroll
      for (int vg = 0; vg < 8; ++vg)
        acc[ma][na][vg] = fmaf(-2.0f, acc[ma][na][vg], rxv[ma][vg] + ryv[na]);

  // ---- Column argmin (fixed gt y = n, min over pred rows m) -> argx ----
#pragma unroll
  for (int na = 0; na < 4; ++na) {
    int n = n0 + na * 16 + lr;
    float best = acc[0][na][0];
    int  bm = m0 + lh * 8;
#pragma unroll
    for (int ma = 0; ma < 2; ++ma)
#pragma unroll
      for (int vg = 0; vg < 8; ++vg) {
        if (ma == 0 && vg == 0) continue;
        float v = acc[ma][na][vg];
        int   m = m0 + ma * 16 + lh * 8 + vg;
        if (v < best || (v == best && m < bm)) { best = v; bm = m; }
      }
    unsigned long long p = packKey(best, bm);
    unsigned long long q = shflXorU64(p, 16);   // merge the two 8-row halves
    if (q < p) p = q;
    if (lh == 0 && n < ny)
      atomicMin(&argx[(size_t)b * ny + n], p);
  }

  // ---- Row argmin (fixed pred x = m, min over gt cols n) -> argy ----
#pragma unroll
  for (int ma = 0; ma < 2; ++ma)
#pragma unroll
    for (int vg = 0; vg < 8; ++vg) {
      int m = m0 + ma * 16 + lh * 8 + vg;
      float best = acc[ma][0][vg];
      int  bn = n0 + lr;
#pragma unroll
      for (int na = 1; na < 4; ++na) {
        float v = acc[ma][na][vg];
        int   n = n0 + na * 16 + lr;
        if (v < best) { best = v; bn = n; }
      }
      unsigned long long p = packKey(best, bn);
#pragma unroll
      for (int off = 1; off < 16; off <<= 1) {   // butterfly within the half
        unsigned long long q = shflXorU64(p, off);
        if (q < p) p = q;
      }
      if (lr == 0 && m < nx)
        atomicMin(&argy[(size_t)b * nx + m], p);
    }
}

// ---------------------------------------------------------------------------
// Kernel 3: KL at matched indices, one wave per point (lane == feature).
// out[b] += 0.5 * (-0.5 * sum_d t) = -0.25 * sum_d t
// ---------------------------------------------------------------------------
__global__ void kl_kernel(const float* __restrict__ mu_p,
                          const float* __restrict__ lv_p,
                          const float* __restrict__ mu_g,
                          const float* __restrict__ lv_g,
                          const unsigned long long* __restrict__ argx,
                          const unsigned long long* __restrict__ argy,
                          float* __restrict__ out,
                          int bs, int nx, int ny) {
  int wid  = (blockIdx.x * blockDim.x + threadIdx.x) >> 5;
  int lane = threadIdx.x & 31;
  int t1 = bs * ny;                 // loss_1 points
  int tot = t1 + bs * nx;          // + loss_2 points
  if (wid >= tot) return;

  int b, qidx, oidx;               // q-params from preds, o-params from gts
  if (wid < t1) {                  // loss_1: gt y matched to pred argx[y]
    b = wid / ny;
    oidx = wid - b * ny;
    qidx = (int)(argx[wid] & 0xffffffffu);
  } else {                         // loss_2: pred x matched to gt argy[x]
    int j = wid - t1;
    b = j / nx;
    qidx = j - b * nx;
    oidx = (int)(argy[j] & 0xffffffffu);
  }

  size_t qo = ((size_t)b * nx + qidx) * DF + lane;
  size_t oo = ((size_t)b * ny + oidx) * DF + lane;
  float lv  = lv_p[qo];
  float mu  = mu_p[qo];
  float lvo = lv_g[oo];
  float muo = mu_g[oo];
  float dm  = mu - muo;
  float t = 1.0f + lv - lvo - __expf(lv - lvo) - dm * dm * __expf(-lvo);

#pragma unroll
  for (int off = 16; off > 0; off >>= 1) t += __shfl_xor(t, off, 32);
  if (lane == 0) atomicAdd(&out[b], -0.25f * t);
}

// ---------------------------------------------------------------------------
extern "C" void kernel_launch(void* const* d_in, const int* in_sizes, int n_in,
                              void* d_out, int out_size, void* d_ws, size_t ws_size,
                              hipStream_t stream) {
  const float* mu_p = (const float*)d_in[0];
  const float* lv_p = (const float*)d_in[1];
  const float* mu_g = (const float*)d_in[2];
  const float* lv_g = (const float*)d_in[3];
  float* out = (float*)d_out;

  const int bs = out_size;                     // reference: 8
  const int nx = in_sizes[0] / (bs * DF);      // 4096
  const int ny = in_sizes[2] / (bs * DF);      // 4096

  // Workspace layout: [argx u64 bs*ny][argy u64 bs*nx][rx f32 bs*nx][ry f32 bs*ny]
  unsigned long long* argx = (unsigned long long*)d_ws;
  unsigned long long* argy = argx + (size_t)bs * ny;
  float* rx = (float*)(argy + (size_t)bs * nx);
  float* ry = rx + (size_t)bs * nx;

  int nInit = bs * (nx > ny ? nx : ny);
  init_kernel<<<(nInit + 255) / 256, 256, 0, stream>>>(argx, argy, out,
                                                       bs * ny, bs * nx, bs);

  long long nwaves = (long long)bs * (nx + ny);
  int nblk = (int)((nwaves * 32 + 255) / 256);
  norm_kernel<<<nblk, 256, 0, stream>>>(mu_p, mu_g, rx, ry, bs, nx, ny);

  dim3 grid((nx + TILE_M - 1) / TILE_M, (ny + TILE_N - 1) / TILE_N, bs);
  chamfer_argmin_wmma<<<grid, 256, 0, stream>>>(mu_p, mu_g, rx, ry,
                                                argx, argy, nx, ny);

  kl_kernel<<<nblk, 256, 0, stream>>>(mu_p, lv_p, mu_g, lv_g,
                                      argx, argy, out, bs, nx, ny);
}